// TrajsFeaturesSimple_39152921870523
// MI455X (gfx1250) — compile-verified
//
#include <hip/hip_runtime.h>
#include <math.h>

typedef float v2f __attribute__((ext_vector_type(2)));
typedef float v8f __attribute__((ext_vector_type(8)));

#define WAVE 32
#define NEG_INF (-3.402823466e38f)

__device__ __forceinline__ float wave_sum(float v) {
    #pragma unroll
    for (int o = 16; o > 0; o >>= 1) v += __shfl_xor(v, o, WAVE);
    return v;
}
__device__ __forceinline__ float wave_max(float v) {
    #pragma unroll
    for (int o = 16; o > 0; o >>= 1) v = fmaxf(v, __shfl_xor(v, o, WAVE));
    return v;
}

// In-place inclusive cumsum of s[0..511] (zero-padded past T); result stored to
// out[base+i] for i < T.  Full f32 precision on the matrix unit:
// a 16x16 tile times an upper-triangular ones matrix == row-wise prefix sums.
// K=16 is a chain of four V_WMMA_F32_16X16X4_F32 (K=4) ops; a wave32 shuffle
// scan stitches the 32 sixteen-element chunks. One wave per block -> EXEC all 1s.
__device__ __forceinline__ void wmma_cumsum512(float* s, float* out, int base, int T)
{
    const int lane = threadIdx.x;
    const int half = lane >> 4;   // lanes 16-31 hold K=2,3 of each A slice
    const int l15  = lane & 15;
#if __has_builtin(__builtin_amdgcn_wmma_f32_16x16x4_f32)
    #pragma unroll
    for (int t = 0; t < 2; ++t) {
        const int tb = 256 * t;
        v2f a[4];
        #pragma unroll
        for (int j = 0; j < 4; ++j) {                 // A: 16x16 tile, rows=chunks
            const int k0 = 4 * j + 2 * half;
            a[j].x = s[tb + 16 * l15 + k0 + 0];
            a[j].y = s[tb + 16 * l15 + k0 + 1];
        }
        __syncthreads();                              // all reads before in-place writes
        v8f c = {0.f,0.f,0.f,0.f,0.f,0.f,0.f,0.f};
        #pragma unroll
        for (int j = 0; j < 4; ++j) {                 // B: triangular ones, U[k][n]=k<=n
            const int k0 = 4 * j + 2 * half;
            v2f b;
            b.x = (k0 + 0 <= l15) ? 1.f : 0.f;
            b.y = (k0 + 1 <= l15) ? 1.f : 0.f;
            c = __builtin_amdgcn_wmma_f32_16x16x4_f32(false, a[j], false, b,
                                                      (short)0, c, false, false);
        }
        #pragma unroll
        for (int v = 0; v < 8; ++v) {                 // D layout: lane holds col l15, rows v+8*half
            const int M = v + 8 * half;
            s[tb + 16 * M + l15] = c[v];
        }
        __syncthreads();
    }
#else
    #pragma unroll 1
    for (int ch = lane; ch < 32; ch += WAVE) {        // scalar fallback
        float run = 0.f;
        for (int k = 0; k < 16; ++k) { run += s[16*ch+k]; s[16*ch+k] = run; }
    }
    __syncthreads();
#endif
    // stitch the 32 chunk totals with a wave32 shuffle scan
    float incl = s[16 * lane + 15];
    #pragma unroll
    for (int off = 1; off < 32; off <<= 1) {
        float u = __shfl_up(incl, off, WAVE);
        if (lane >= off) incl += u;
    }
    float excl = __shfl_up(incl, 1, WAVE);
    if (lane == 0) excl = 0.f;
    __shared__ float offs[WAVE];
    offs[lane] = excl;
    __syncthreads();
    for (int i = lane; i < T; i += WAVE)
        out[base + i] = s[i] + offs[i >> 4];
    __syncthreads();
}

// Pass 1: one wave per graph. Streams P/time once, produces all per-graph
// scalars + per-node intermediates (dr, cumsums via WMMA, local cummax).
__global__ __launch_bounds__(WAVE)
void k1_graph(const float* __restrict__ P, const float* __restrict__ tme,
              int N, int G, int T,
              float* __restrict__ cum_d, float* __restrict__ cum_sd, float* __restrict__ cum_qd,
              float* __restrict__ drx, float* __restrict__ dry, float* __restrict__ lcmax,
              float* __restrict__ g_dur, float* __restrict__ g_std,
              float* __restrict__ g_totd, float* __restrict__ g_totsd, float* __restrict__ g_totqd,
              float* __restrict__ g_max, float* __restrict__ g_first,
              float* __restrict__ scales)
{
    __shared__ float sD[512], sS[512], sQ[512], sO[512];
    const int lane = threadIdx.x;
    for (int g = blockIdx.x; g < G; g += gridDim.x) {
        const int base = g * T;
        for (int i = lane; i < 512; i += WAVE) { sD[i]=0.f; sS[i]=0.f; sQ[i]=0.f; sO[i]=0.f; }
        __syncthreads();

        float ssum=0.f, s2=0.f, totd=0.f, totsd=0.f, totqd=0.f;
        float px=0.f, px2=0.f, py=0.f, py2=0.f, tmax=NEG_INF;
        for (int i = lane; i < T; i += WAVE) {
            const float2 p = ((const float2*)P)[base + i];
            const bool last = (i == T - 1);
            float dx = 0.f, dy = 0.f;
            if (!last) {
                const float2 pn = ((const float2*)P)[base + i + 1];
                dx = pn.x - p.x; dy = pn.y - p.y;
            }
            const float dn  = sqrtf(1e-5f + dx*dx + dy*dy);
            const float dn2 = dn * dn, dn4 = dn2 * dn2;
            sD[i] = dn; sS[i] = dn2; sQ[i] = dn4;
            const float dist = sqrtf(p.x*p.x + p.y*p.y + 1e-7f);
            sO[i] = dist;
            drx[base+i] = dx; dry[base+i] = dy;
            if (!last) { ssum += dn; s2 += dn2; }
            totd += dn; totsd += dn2; totqd += dn4;
            px += p.x; px2 += p.x*p.x; py += p.y; py2 += p.y*p.y;
            tmax = fmaxf(tmax, tme[base+i]);
            if (i == 0) g_first[g] = dist;
        }
        __syncthreads();
        ssum = wave_sum(ssum); s2 = wave_sum(s2);
        totd = wave_sum(totd); totsd = wave_sum(totsd); totqd = wave_sum(totqd);
        px = wave_sum(px); px2 = wave_sum(px2); py = wave_sum(py); py2 = wave_sum(py2);
        tmax = wave_max(tmax);

        const float cnt  = (float)(T - 1);
        const float mean = ssum / cnt;
        const float evar = s2 / cnt;                 // reference's "step_var" = E[x^2]
        const float sstd = sqrtf(evar - mean*mean);
        const float Tf   = (float)T;
        const float pos_std = sqrtf((px2/Tf - (px/Tf)*(px/Tf)) + (py2/Tf - (py/Tf)*(py/Tf)));
        if (lane == 0) {
            g_dur[g]=tmax; g_std[g]=sstd;
            g_totd[g]=totd; g_totsd[g]=totsd; g_totqd[g]=totqd;
            // scales dict in JAX pytree (sorted-key) order
            scales[0*G+g] = tmax / Tf;   // mean_time_step
            scales[1*G+g] = pos_std;     // pos_std
            scales[2*G+g] = mean;        // step_mean
            scales[3*G+g] = sstd;        // step_std
            scales[4*G+g] = ssum;        // step_sum
            scales[5*G+g] = evar;        // step_var
        }

        wmma_cumsum512(sD, cum_d,  base, T);
        wmma_cumsum512(sS, cum_sd, base, T);
        wmma_cumsum512(sQ, cum_qd, base, T);

        // per-graph inclusive cummax of dist-to-origin (lane-chunk + shuffle max-scan)
        float run = NEG_INF;
        #pragma unroll 1
        for (int k = 0; k < 16; ++k) {
            run = fmaxf(run, sO[16*lane + k]);
            sO[16*lane + k] = run;
        }
        float incl = run;
        #pragma unroll
        for (int off = 1; off < 32; off <<= 1) {
            float u = __shfl_up(incl, off, WAVE);
            if (lane >= off) incl = fmaxf(incl, u);
        }
        float excl = __shfl_up(incl, 1, WAVE);
        if (lane == 0) excl = NEG_INF;
        for (int k = 0; k < 16; ++k) {
            const int i = 16*lane + k;
            if (i < T) lcmax[base+i] = fmaxf(sO[i], excl);
        }
        const float gm = __shfl(incl, 31, WAVE);
        if (lane == 0) g_max[g] = gm;
        __syncthreads();
    }
}

// Pass 2: running max over graph maxima (replicates the reference's GLOBAL cummax).
__global__ __launch_bounds__(1024)
void k2_prefmax(const float* __restrict__ gmax, float* __restrict__ pref, int G)
{
    __shared__ float sm[1024];
    __shared__ float carry;
    const int t = threadIdx.x;
    if (t == 0) carry = NEG_INF;
    __syncthreads();
    for (int b = 0; b < G; b += 1024) {
        const int idx = b + t;
        sm[t] = (idx < G) ? gmax[idx] : NEG_INF;
        __syncthreads();
        for (int off = 1; off < 1024; off <<= 1) {
            const float u = (t >= off) ? sm[t - off] : NEG_INF;
            __syncthreads();
            sm[t] = fmaxf(sm[t], u);
            __syncthreads();
        }
        const float c = carry;
        if (idx < G) pref[idx] = (t == 0) ? c : fmaxf(c, sm[t - 1]);
        __syncthreads();
        if (t == 0) carry = fmaxf(c, sm[1023]);
        __syncthreads();
    }
}

// Pass 3: node features X (N x 6).
__global__ void k3_nodes(const float* __restrict__ P, const float* __restrict__ tme,
                         const float* __restrict__ cum_d, const float* __restrict__ cum_sd,
                         const float* __restrict__ cum_qd, const float* __restrict__ lcmax,
                         const float* __restrict__ g_dur, const float* __restrict__ g_std,
                         const float* __restrict__ g_totd, const float* __restrict__ g_totsd,
                         const float* __restrict__ g_totqd, const float* __restrict__ g_first,
                         const float* __restrict__ pref,
                         int N, int T, float* __restrict__ Xout)
{
    const int i = blockIdx.x * blockDim.x + threadIdx.x;
    if (i >= N) return;
    const int g = i / T;
    const float t  = tme[i];
    const float tn = t / g_dur[g];
    const bool  nm = (tn == 0.f);
    const float tns   = nm ? 1.f : tn;
    const float tsafe = nm ? 1.f : t;
    const float2 p = ((const float2*)P)[i];
    const float dist = sqrtf(p.x*p.x + p.y*p.y + 1e-7f);
    const float mdo  = fmaxf(lcmax[i], pref[g]) + g_first[g];  // global cummax + first val
    const float inv  = 1.f / (g_std[g] * sqrtf(tsafe));
    float x0 = tn;
    float x1 = cum_d[i]  / (g_totd[g]  * tns);
    float x2 = cum_sd[i] / (g_totsd[g] * tns);
    float x3 = cum_qd[i] / (g_totqd[g] * tns);
    float x4 = dist * inv;
    float x5 = mdo  * inv;
    if (nm) { x0=x1=x2=x3=x4=x5=0.f; }
    float* row = Xout + (size_t)i * 6;
    row[0]=x0; row[1]=x1; row[2]=x2; row[3]=x3; row[4]=x4; row[5]=x5;
}

// Pass 4: edge features E (Ecnt x 6), pure gathers.
__global__ void k4_edges(const float* __restrict__ tme, const float* __restrict__ P,
                         const int* __restrict__ src, const int* __restrict__ tgt,
                         const float* __restrict__ cum_d, const float* __restrict__ cum_sd,
                         const float* __restrict__ cum_qd,
                         const float* __restrict__ drx, const float* __restrict__ dry,
                         const float* __restrict__ g_dur, const float* __restrict__ g_std,
                         const float* __restrict__ g_totd, const float* __restrict__ g_totsd,
                         const float* __restrict__ g_totqd,
                         int T, int Ecnt, float* __restrict__ Eout)
{
    const int e = blockIdx.x * blockDim.x + threadIdx.x;
    if (e >= Ecnt) return;
    const int s = src[e], t = tgt[e];
    const int gs = s / T, gt = t / T;
    const float ts = tme[s], tt = tme[t];
    const float td = tt - ts;
    const bool  em = (td == 0.0f);
    const float tdn = tt / g_dur[gt] - ts / g_dur[gs];
    const float tdn_safe = em ? 1.f : fabsf(tdn);
    const float td_safe  = em ? 1.f : fabsf(td);
    const float2 pt = ((const float2*)P)[t];
    const float2 ps = ((const float2*)P)[s];
    const float ddx = pt.x - ps.x, ddy = pt.y - ps.y;
    const float d = sqrtf(em ? 1.f : (ddx*ddx + ddy*ddy));
    const float stdt = g_std[gt], stds = g_std[gs];
    const float corr = (drx[t]*drx[s] + dry[t]*dry[s]) / (stdt * stds);
    // NOTE: reference precedence is a - b/(c*d) — replicated exactly.
    float o0 = td;
    float o1 = cum_d[t]  - cum_d[s]  / (g_totd[gt]  * tdn_safe);
    float o2 = cum_sd[t] - cum_sd[s] / (g_totsd[gt] * tdn_safe);
    float o3 = cum_qd[t] - cum_qd[s] / (g_totqd[gt] * tdn_safe);
    float o4 = d / (stdt * sqrtf(td_safe));
    float o5 = corr;
    if (em) { o0=o1=o2=o3=o4=o5=0.f; }
    float* row = Eout + (size_t)e * 6;
    row[0]=o0; row[1]=o1; row[2]=o2; row[3]=o3; row[4]=o4; row[5]=o5;
}

extern "C" void kernel_launch(void* const* d_in, const int* in_sizes, int n_in,
                              void* d_out, int out_size, void* d_ws, size_t ws_size,
                              hipStream_t stream)
{
    (void)n_in; (void)ws_size;
    const float* P   = (const float*)d_in[0];
    const float* tme = (const float*)d_in[1];
    const int*   src = (const int*)d_in[3];
    const int*   tgt = (const int*)d_in[4];
    const int N    = in_sizes[1];
    const int Ecnt = in_sizes[3];
    // out = X(6N) + E(6Ecnt) + 6 per-graph arrays -> recover G on the host.
    const int G = (int)(((long long)out_size - 6LL*N - 6LL*Ecnt) / 6LL);
    const int T = (G > 0) ? (N / G) : 1;

    float* ws = (float*)d_ws;
    float* cum_d  = ws;
    float* cum_sd = ws + (size_t)N;
    float* cum_qd = ws + 2*(size_t)N;
    float* drx    = ws + 3*(size_t)N;
    float* dry    = ws + 4*(size_t)N;
    float* lcmax  = ws + 5*(size_t)N;
    float* pg     = ws + 6*(size_t)N;
    float* g_dur   = pg;
    float* g_std   = pg + (size_t)G;
    float* g_totd  = pg + 2*(size_t)G;
    float* g_totsd = pg + 3*(size_t)G;
    float* g_totqd = pg + 4*(size_t)G;
    float* g_max   = pg + 5*(size_t)G;
    float* g_first = pg + 6*(size_t)G;
    float* prefm   = pg + 7*(size_t)G;

    float* Xout = (float*)d_out;
    float* Eout = Xout + 6*(size_t)N;
    float* Sout = Eout + 6*(size_t)Ecnt;

    k1_graph<<<G, WAVE, 0, stream>>>(P, tme, N, G, T,
        cum_d, cum_sd, cum_qd, drx, dry, lcmax,
        g_dur, g_std, g_totd, g_totsd, g_totqd, g_max, g_first, Sout);
    k2_prefmax<<<1, 1024, 0, stream>>>(g_max, prefm, G);
    k3_nodes<<<(N + 255) / 256, 256, 0, stream>>>(P, tme, cum_d, cum_sd, cum_qd, lcmax,
        g_dur, g_std, g_totd, g_totsd, g_totqd, g_first, prefm, N, T, Xout);
    k4_edges<<<(Ecnt + 255) / 256, 256, 0, stream>>>(tme, P, src, tgt,
        cum_d, cum_sd, cum_qd, drx, dry,
        g_dur, g_std, g_totd, g_totsd, g_totqd, T, Ecnt, Eout);
}